// GaussianBlur3D_68556267978927
// MI455X (gfx1250) — compile-verified
//
#include <hip/hip_runtime.h>
#include <stdint.h>

// Fused separable 3D "Gaussian" blur for MI455X (gfx1250, wave32).
//   pass W : 5-tap weights exp(-c^2)/Sb      -> V_WMMA_F32_16X16X4_F32
//            (banded Toeplitz matmul, 5 chained K=4 WMMAs per 16x16 tile)
//   pass H : 5-tap uniform (1/5)             -> VALU, sliding 11-row window
//   pass D : 5-tap weights exp(-c^2/2)/Sa    -> register ring along D
// Input planes staged into double-buffered LDS with global_load_async_to_lds_b64
// (ASYNCcnt) so the next plane's DMA overlaps this plane's compute.

typedef __attribute__((ext_vector_type(2))) float v2f;
typedef __attribute__((ext_vector_type(8))) float v8f;

#define ND     2
#define DD     192
#define HH     256
#define WW     256
#define HT     28      // output rows per tile (HT+4 = 32 = 2 WMMA row-tiles)
#define ROWS   32      // HT + 4 halo rows
#define WT     64      // output cols per tile
#define PITCH  68      // LDS row pitch in floats (breaks 64-bank alignment)
#define DC     32      // D-chunk per workgroup
#define HTILES 10      // ceil(256/28)
#define WTILES 4

__device__ __forceinline__ void load_plane_async(
    const float* __restrict__ x, float* lds, int bn, int p, int h0, int w0, int tid)
{
  const bool pin = (p >= 0) && (p < DD);
  const float* src = x + ((size_t)bn * DD + (size_t)(pin ? p : 0)) * ((size_t)HH * WW);
  // 32 rows x 34 float2-pairs per plane tile. w0 is a multiple of 64 and the
  // halo shift is -2, so every 8-byte pair is entirely in or entirely out of
  // bounds -> either one async b64 DMA into LDS or a zero fill.
#pragma unroll 1
  for (int slot = tid; slot < ROWS * 34; slot += 256) {
    const int row = slot / 34;
    const int pr  = slot - row * 34;
    const int gh  = h0 - 2 + row;
    const int gw  = w0 - 2 + pr * 2;
    float* ldst = lds + row * PITCH + pr * 2;
    if (pin && (gh >= 0) && (gh < HH) && (gw >= 0) && (gw < WW)) {
      const uint32_t laddr = (uint32_t)(uintptr_t)ldst;              // LDS byte addr
      const uint64_t gaddr = (uint64_t)(uintptr_t)(src + (size_t)gh * WW + gw);
      asm volatile("global_load_async_to_lds_b64 %0, %1, off"
                   :: "v"(laddr), "v"(gaddr) : "memory");
    } else {
      ldst[0] = 0.0f;
      ldst[1] = 0.0f;
    }
  }
}

__global__ __launch_bounds__(256)
void GaussianBlur3D_68556267978927_kernel(const float* __restrict__ x,
                                          float* __restrict__ out)
{
  __shared__ __align__(16) float s_in[2][ROWS * PITCH];  // double-buffered input planes
  __shared__ __align__(16) float s_row[ROWS * PITCH];    // W-filtered plane

  const int tid = threadIdx.x;
  const int bn  = blockIdx.z;
  const int d0  = blockIdx.y * DC;
  const int h0  = (blockIdx.x % HTILES) * HT;
  const int w0  = (blockIdx.x / HTILES) * WT;

  // ---- separable weights (sigma = 1) -------------------------------------
  const float e05 = __expf(-0.5f), e2 = __expf(-2.0f);
  const float e1  = __expf(-1.0f), e4 = __expf(-4.0f);
  const float sa  = 1.0f + 2.0f * e05 + 2.0f * e2;   // D-axis norm
  const float sb  = 1.0f + 2.0f * e1  + 2.0f * e4;   // W-axis norm
  const float wD0 = e2 / sa, wD1 = e05 / sa, wD2 = 1.0f / sa;
  const float tv[5] = { e4 / sb, e1 / sb, 1.0f / sb, e1 / sb, e4 / sb };

  // ---- per-lane WMMA operand prep ----------------------------------------
  const int lane  = tid & 31;
  const int wave  = tid >> 5;
  const int m     = lane & 15;          // A: row M / D: column N
  const int hiA   = lane >> 4;          // half-wave selects K pair
  const int rbase = (wave >> 2) << 4;   // {0,16}
  const int wbase = (wave & 3) << 4;    // {0,16,32,48}

  // Constant banded-Toeplitz B blocks: B_s[K][n] = tv[4s + K - n] (zero off band).
  // f32 B layout: component v of Bm[s] holds K = v + 2*hiA, column n = lane&15.
  v2f Bm[5];
#pragma unroll
  for (int s = 0; s < 5; ++s) {
    const int base = 4 * s + 2 * hiA - m;
    float bx = 0.0f, by = 0.0f;
#pragma unroll
    for (int t = 0; t < 5; ++t) {
      bx = (base == t)     ? tv[t] : bx;
      by = (base + 1 == t) ? tv[t] : by;
    }
    Bm[s].x = bx;
    Bm[s].y = by;
  }

  // ---- H/D-pass thread mapping & register ring ---------------------------
  // Thread owns 7 CONSECUTIVE output rows (g*7 .. g*7+6) at one column, so the
  // 7 box-filter sums share a sliding 11-row window: 11 LDS loads, not 35.
  const int wcol = tid & 63;            // output column in tile
  const int g    = tid >> 6;            // 0..3 -> rows g*7 .. g*7+6
  float rng[7][5];
#pragma unroll
  for (int j = 0; j < 7; ++j)
#pragma unroll
    for (int t = 0; t < 5; ++t) rng[j][t] = 0.0f;

  // ---- prologue: DMA first plane -----------------------------------------
  load_plane_async(x, &s_in[0][0], bn, d0 - 2, h0, w0, tid);
  asm volatile("s_wait_asynccnt 0" ::: "memory");
  __syncthreads();

#pragma unroll 1
  for (int p = d0 - 2; p <= d0 + DC + 1; ++p) {
    const int cur = (p + 2) & 1;

    // Kick off the DMA for the next plane (overlaps with compute below).
    if (p < d0 + DC + 1)
      load_plane_async(x, &s_in[cur ^ 1][0], bn, p + 1, h0, w0, tid);

    // ---- W pass: 16x16 output tile per wave = A(16x20) x Toeplitz(20x16) --
    {
      const float* inb = &s_in[cur][0];
      v8f acc = {0.0f, 0.0f, 0.0f, 0.0f, 0.0f, 0.0f, 0.0f, 0.0f};
#pragma unroll
      for (int s = 0; s < 5; ++s) {
        const v2f a = *(const v2f*)(inb + (rbase + m) * PITCH + wbase + 4 * s + 2 * hiA);
        acc = __builtin_amdgcn_wmma_f32_16x16x4_f32(
            false, a, false, Bm[s], (short)0, acc, false, false);
      }
#pragma unroll
      for (int r = 0; r < 8; ++r)
        s_row[(rbase + r + 8 * hiA) * PITCH + wbase + m] = acc[r];
    }
    __syncthreads();

    // ---- H pass (uniform 1/5): sliding 11-row window -----------------------
    const float* rp = &s_row[(g * 7) * PITCH + wcol];
    float t0, t1, t2, t3, t4, t5, t6, t7, t8, t9, t10;
    t0  = rp[0 * PITCH];  t1 = rp[1 * PITCH];  t2 = rp[2 * PITCH];
    t3  = rp[3 * PITCH];  t4 = rp[4 * PITCH];  t5 = rp[5 * PITCH];
    t6  = rp[6 * PITCH];  t7 = rp[7 * PITCH];  t8 = rp[8 * PITCH];
    t9  = rp[9 * PITCH];  t10 = rp[10 * PITCH];
    // pairwise partials a[j] = t[j] + t[j+1]; sum5[j] = a[j] + a[j+2] + t[j+4]
    const float a0 = t0 + t1, a1 = t1 + t2, a2 = t2 + t3, a3 = t3 + t4;
    const float a4 = t4 + t5, a5 = t5 + t6, a6 = t6 + t7, a7 = t7 + t8;
    const float a8 = t8 + t9;
    float hf[7];
    hf[0] = 0.2f * (a0 + a2 + t4);
    hf[1] = 0.2f * (a1 + a3 + t5);
    hf[2] = 0.2f * (a2 + a4 + t6);
    hf[3] = 0.2f * (a3 + a5 + t7);
    hf[4] = 0.2f * (a4 + a6 + t8);
    hf[5] = 0.2f * (a5 + a7 + t9);
    hf[6] = 0.2f * (a6 + a8 + t10);

    // ---- ring shift + D pass ------------------------------------------------
#pragma unroll
    for (int j = 0; j < 7; ++j) {
      rng[j][0] = rng[j][1]; rng[j][1] = rng[j][2]; rng[j][2] = rng[j][3];
      rng[j][3] = rng[j][4]; rng[j][4] = hf[j];
    }
    if (p >= d0 + 2) {
      const int q = p - 2;                   // output depth plane
#pragma unroll
      for (int j = 0; j < 7; ++j) {
        const int h = h0 + g * 7 + j;
        if (h < HH) {
          const float o = wD0 * (rng[j][0] + rng[j][4]) +
                          wD1 * (rng[j][1] + rng[j][3]) +
                          wD2 *  rng[j][2];
          out[(((size_t)bn * DD + q) * HH + h) * (size_t)WW + (w0 + wcol)] = o;
        }
      }
    }

    // Next plane's DMA must be complete before anyone touches the other buffer.
    asm volatile("s_wait_asynccnt 0" ::: "memory");
    __syncthreads();
  }
}

extern "C" void kernel_launch(void* const* d_in, const int* in_sizes, int n_in,
                              void* d_out, int out_size, void* d_ws, size_t ws_size,
                              hipStream_t stream) {
  (void)in_sizes; (void)n_in; (void)out_size; (void)d_ws; (void)ws_size;
  const float* xin = (const float*)d_in[0];   // (2,1,192,256,256) f32
  float* o = (float*)d_out;                   // same shape, f32
  dim3 grid(WTILES * HTILES, DD / DC, ND);    // (40, 6, 2)
  dim3 block(256);
  GaussianBlur3D_68556267978927_kernel<<<grid, block, 0, stream>>>(xin, o);
}